// LinearAttention_37873021616336
// MI455X (gfx1250) — compile-verified
//
#include <hip/hip_runtime.h>
#include <math.h>

typedef float v2f __attribute__((ext_vector_type(2)));
typedef float v8f __attribute__((ext_vector_type(8)));

#define B_    8
#define C_    512
#define N_    1024          // H*W
#define C3_   1536
#define GRP_  8
#define GCH_  64            // channels per group
#define GELT_ (GCH_ * N_)   // elements per (b,g) group = 65536
#define EPS_  1e-5f

// ---------------------------------------------------------------------------
// Weight transpose: out[c][r] = in[r][c]   (R x Cc) -> (Cc x R)
// ---------------------------------------------------------------------------
__global__ __launch_bounds__(256) void kT_transpose(const float* __restrict__ in,
                                                    float* __restrict__ out,
                                                    int R, int Cc) {
  __shared__ float tile[32][33];
  int c0 = blockIdx.x * 32, r0 = blockIdx.y * 32;
  int tx = threadIdx.x, ty = threadIdx.y;            // 32 x 8
  #pragma unroll
  for (int i = 0; i < 32; i += 8)
    tile[ty + i][tx] = in[(size_t)(r0 + ty + i) * Cc + c0 + tx];
  __syncthreads();
  #pragma unroll
  for (int i = 0; i < 32; i += 8)
    out[(size_t)(c0 + ty + i) * R + r0 + tx] = tile[tx][ty + i];
}

// ---------------------------------------------------------------------------
// Batched transpose of the q slice of qkv: qT[b][d][n] = qkv[b][n][d]
// coalesced on both sides via LDS tile. Grid (C/32, N/32, B), block (32,8)
// ---------------------------------------------------------------------------
__global__ __launch_bounds__(256) void kT_qtranspose(const float* __restrict__ qkv,
                                                     float* __restrict__ qT) {
  __shared__ float tile[32][33];
  int d0 = blockIdx.x * 32, n0 = blockIdx.y * 32;
  const float* in = qkv + (size_t)blockIdx.z * N_ * C3_;
  float*      out = qT  + (size_t)blockIdx.z * C_ * N_;
  int tx = threadIdx.x, ty = threadIdx.y;
  #pragma unroll
  for (int i = 0; i < 32; i += 8)
    tile[ty + i][tx] = in[(size_t)(n0 + ty + i) * C3_ + d0 + tx];
  __syncthreads();
  #pragma unroll
  for (int i = 0; i < 32; i += 8)
    out[(size_t)(d0 + ty + i) * N_ + n0 + tx] = tile[tx][ty + i];
}

// ---------------------------------------------------------------------------
// GroupNorm stats: one block per (b,g); group slice is contiguous 64K floats (256KB)
// ---------------------------------------------------------------------------
__global__ __launch_bounds__(256) void kT_gn_stats(const float* __restrict__ x,
                                                   float* __restrict__ mean,
                                                   float* __restrict__ rstd) {
  __shared__ float s1[256], s2[256];
  int bg = blockIdx.x;                               // 0..63
  const float4* p = (const float4*)(x + (size_t)bg * GELT_);
  float s = 0.f, q = 0.f;
  for (int i = threadIdx.x; i < GELT_ / 4; i += 256) {
    float4 v = p[i];
    s += v.x + v.y + v.z + v.w;
    q += v.x * v.x + v.y * v.y + v.z * v.z + v.w * v.w;
  }
  s1[threadIdx.x] = s; s2[threadIdx.x] = q;
  __syncthreads();
  for (int off = 128; off > 0; off >>= 1) {
    if (threadIdx.x < off) {
      s1[threadIdx.x] += s1[threadIdx.x + off];
      s2[threadIdx.x] += s2[threadIdx.x + off];
    }
    __syncthreads();
  }
  if (threadIdx.x == 0) {
    float m = s1[0] * (1.0f / GELT_);
    float v = s2[0] * (1.0f / GELT_) - m * m;
    mean[bg] = m;
    rstd[bg] = rsqrtf(v + EPS_);
  }
}

// ---------------------------------------------------------------------------
// GroupNorm apply: one block per (b,c) row of 1024 floats
// ---------------------------------------------------------------------------
__global__ __launch_bounds__(256) void kT_gn_apply(const float* __restrict__ x,
                                                   const float* __restrict__ gamma,
                                                   const float* __restrict__ beta,
                                                   const float* __restrict__ mean,
                                                   const float* __restrict__ rstd,
                                                   float* __restrict__ xn) {
  int bc = blockIdx.x;                               // b*512 + c
  int c = bc & (C_ - 1);
  int bg = ((bc >> 9) << 3) + (c >> 6);
  float gm = gamma[c] * rstd[bg];
  float bt = beta[c] - mean[bg] * gm;                // xn = x*gm + bt
  const float4* pin  = (const float4*)(x  + (size_t)bc * N_);
  float4*       pout = (float4*)(xn + (size_t)bc * N_);
  float4 v = pin[threadIdx.x];
  v.x = v.x * gm + bt; v.y = v.y * gm + bt;
  v.z = v.z * gm + bt; v.w = v.w * gm + bt;
  pout[threadIdx.x] = v;
}

// ---------------------------------------------------------------------------
// softmax over channels for k (in place): one block per (b,n), 512 elems
// ---------------------------------------------------------------------------
__global__ __launch_bounds__(256) void kT_softmax_k(float* __restrict__ qkv) {
  __shared__ float red[256];
  float* row = qkv + (size_t)blockIdx.x * C3_ + C_;
  int t = threadIdx.x;
  float a = row[t], b = row[t + 256];
  red[t] = fmaxf(a, b);
  __syncthreads();
  for (int off = 128; off > 0; off >>= 1) {
    if (t < off) red[t] = fmaxf(red[t], red[t + off]);
    __syncthreads();
  }
  float mx = red[0];
  __syncthreads();
  float e0 = __expf(a - mx), e1 = __expf(b - mx);
  red[t] = e0 + e1;
  __syncthreads();
  for (int off = 128; off > 0; off >>= 1) {
    if (t < off) red[t] += red[t + off];
    __syncthreads();
  }
  float inv = 1.0f / red[0];
  row[t] = e0 * inv;
  row[t + 256] = e1 * inv;
}

// ---------------------------------------------------------------------------
// softmax over a contiguous row of 1024 floats, in place (for qT rows)
// one block per (b,d); float4 per thread
// ---------------------------------------------------------------------------
__global__ __launch_bounds__(256) void kT_softmax_row1024(float* __restrict__ qT) {
  __shared__ float red[256];
  float4* row = (float4*)(qT + (size_t)blockIdx.x * N_);
  int t = threadIdx.x;
  float4 v = row[t];
  red[t] = fmaxf(fmaxf(v.x, v.y), fmaxf(v.z, v.w));
  __syncthreads();
  for (int off = 128; off > 0; off >>= 1) {
    if (t < off) red[t] = fmaxf(red[t], red[t + off]);
    __syncthreads();
  }
  float mx = red[0];
  __syncthreads();
  float4 e;
  e.x = __expf(v.x - mx); e.y = __expf(v.y - mx);
  e.z = __expf(v.z - mx); e.w = __expf(v.w - mx);
  red[t] = e.x + e.y + e.z + e.w;
  __syncthreads();
  for (int off = 128; off > 0; off >>= 1) {
    if (t < off) red[t] += red[t + off];
    __syncthreads();
  }
  float inv = 1.0f / red[0];
  e.x *= inv; e.y *= inv; e.z *= inv; e.w *= inv;
  row[t] = e;
}

// ---------------------------------------------------------------------------
// Templated f32 WMMA GEMM. Compile-time leading dims/K -> immediate-offset
// loads. Per-wave output tile: 32 (m) x 64 (n): 2 A frags, 4 B frags,
// 8 accumulators -> 8 WMMAs per k-step at 1.5 VMEM loads per WMMA.
//   D[m][n] = sum_k Am[m][k] * Bm[k][n] (+ bias[n]) (+ residual)
//   A element (m,k) at A + k*LDA + m        (m contiguous)
//   B element (k,n) at B + k*LDB + n        (n contiguous)
//   Store: row-major Cb[m*LDC + n]   or transposed Cb[n*LDC + m] (+res)
// Block = 256 threads = 8 waves covering 64(m) x 256(n).
// Grid: (M/64, Ntot/256, batch)
// ---------------------------------------------------------------------------
template <int LDA, int LDB, int LDC, int K, bool TSTORE, bool HAS_BIAS, bool HAS_RES>
__global__ __launch_bounds__(256) void kT_gemm(
    const float* __restrict__ A, long long strideA,
    const float* __restrict__ B, long long strideB,
    float* __restrict__ Cmat,    long long strideC,
    const float* __restrict__ bias,
    const float* __restrict__ resid, long long strideR) {
  int z = blockIdx.z;
  const float* Ab = A + (size_t)z * strideA;
  const float* Bb = B + (size_t)z * strideB;
  float*       Cb = Cmat + (size_t)z * strideC;

  int wave = threadIdx.x >> 5;
  int lane = threadIdx.x & 31;
  int half = lane >> 4;                // 0: K pair {0,1}; 1: K pair {2,3}
  int ml   = lane & 15;
  int m0 = blockIdx.x * 64  + (wave & 1) * 32;   // wave owns rows m0..m0+31
  int n0 = blockIdx.y * 256 + (wave >> 1) * 64;  // wave owns cols n0..n0+63

  v8f acc[2][4];
  {
    float bv[4];
    #pragma unroll
    for (int s = 0; s < 4; ++s) bv[s] = HAS_BIAS ? bias[n0 + 16 * s + ml] : 0.0f;
    #pragma unroll
    for (int mi = 0; mi < 2; ++mi)
      #pragma unroll
      for (int s = 0; s < 4; ++s)
        #pragma unroll
        for (int r = 0; r < 8; ++r) acc[mi][s][r] = bv[s];
  }

  const float* Ap = Ab + (size_t)(2 * half) * LDA + m0 + ml;  // frag1 at +16 (imm)
  const float* Bp = Bb + (size_t)(2 * half) * LDB + n0 + ml;  // sub-tiles at +16/32/48

  #pragma unroll 1
  for (int kk = 0; kk < K; kk += 16) {
    __builtin_prefetch(Ap + (size_t)20 * LDA, 0, 1);          // global_prefetch A stream
    #pragma unroll
    for (int j = 0; j < 4; ++j) {
      const int k0 = 4 * j * LDA, k1 = (4 * j + 1) * LDA;
      const int h0 = 4 * j * LDB, h1 = (4 * j + 1) * LDB;
      v2f a0; a0.x = Ap[k0];      a0.y = Ap[k1];
      v2f a1; a1.x = Ap[k0 + 16]; a1.y = Ap[k1 + 16];
      v2f b0; b0.x = Bp[h0];      b0.y = Bp[h1];
      v2f b1; b1.x = Bp[h0 + 16]; b1.y = Bp[h1 + 16];
      v2f b2; b2.x = Bp[h0 + 32]; b2.y = Bp[h1 + 32];
      v2f b3; b3.x = Bp[h0 + 48]; b3.y = Bp[h1 + 48];
      acc[0][0] = __builtin_amdgcn_wmma_f32_16x16x4_f32(false, a0, false, b0, (short)0, acc[0][0], false, false);
      acc[0][1] = __builtin_amdgcn_wmma_f32_16x16x4_f32(false, a0, false, b1, (short)0, acc[0][1], false, false);
      acc[0][2] = __builtin_amdgcn_wmma_f32_16x16x4_f32(false, a0, false, b2, (short)0, acc[0][2], false, false);
      acc[0][3] = __builtin_amdgcn_wmma_f32_16x16x4_f32(false, a0, false, b3, (short)0, acc[0][3], false, false);
      acc[1][0] = __builtin_amdgcn_wmma_f32_16x16x4_f32(false, a1, false, b0, (short)0, acc[1][0], false, false);
      acc[1][1] = __builtin_amdgcn_wmma_f32_16x16x4_f32(false, a1, false, b1, (short)0, acc[1][1], false, false);
      acc[1][2] = __builtin_amdgcn_wmma_f32_16x16x4_f32(false, a1, false, b2, (short)0, acc[1][2], false, false);
      acc[1][3] = __builtin_amdgcn_wmma_f32_16x16x4_f32(false, a1, false, b3, (short)0, acc[1][3], false, false);
    }
    Ap += 16 * LDA;
    Bp += 16 * LDB;
  }

  if (!TSTORE) {
    #pragma unroll
    for (int mi = 0; mi < 2; ++mi) {
      #pragma unroll
      for (int r = 0; r < 8; ++r) {
        int m = m0 + 16 * mi + r + 8 * half;
        #pragma unroll
        for (int s = 0; s < 4; ++s)
          Cb[(size_t)m * LDC + n0 + 16 * s + ml] = acc[mi][s][r];
      }
    }
  } else {
    const float* Rb = HAS_RES ? (resid + (size_t)z * strideR) : nullptr;
    #pragma unroll
    for (int mi = 0; mi < 2; ++mi) {
      #pragma unroll
      for (int r = 0; r < 8; ++r) {
        int m = m0 + 16 * mi + r + 8 * half;
        #pragma unroll
        for (int s = 0; s < 4; ++s) {
          size_t i = (size_t)(n0 + 16 * s + ml) * LDC + m;
          float v = acc[mi][s][r];
          if (HAS_RES) v += Rb[i];
          Cb[i] = v;
        }
      }
    }
  }
}

// ---------------------------------------------------------------------------
// Host-side launch
// ---------------------------------------------------------------------------
extern "C" void kernel_launch(void* const* d_in, const int* in_sizes, int n_in,
                              void* d_out, int out_size, void* d_ws, size_t ws_size,
                              hipStream_t stream) {
  const float* x      = (const float*)d_in[0];   // [8,512,32,32]
  const float* gn_w   = (const float*)d_in[1];   // [512]
  const float* gn_b   = (const float*)d_in[2];   // [512]
  const float* qkv_w  = (const float*)d_in[3];   // [1536,512]
  const float* qkv_b  = (const float*)d_in[4];   // [1536]
  const float* proj_w = (const float*)d_in[5];   // [512,512]
  const float* proj_b = (const float*)d_in[6];   // [512]
  float* out = (float*)d_out;                    // [8,512,32,32]

  float* ws = (float*)d_ws;
  // workspace layout (floats)
  const size_t XN_SZ  = (size_t)B_ * C_ * N_;          // 4,194,304
  const size_t QKV_SZ = (size_t)B_ * N_ * C3_;         // 12,582,912
  const size_t CTX_SZ = (size_t)B_ * C_ * C_;          // 2,097,152
  float* mean   = ws;                                  // 64
  float* rstd   = ws + 64;                             // 64
  float* xn     = ws + 256;                            // [B][C][N]
  float* qkv    = xn   + XN_SZ;                        // [B][N][3C]
  float* qT     = qkv  + QKV_SZ;                       // [B][C][N]
  float* ctx    = qT   + XN_SZ;                        // [B][C][C]
  float* o2T    = ctx  + CTX_SZ;                       // [B][C][N]
  float* wqkvT  = o2T  + XN_SZ;                        // [C][3C]
  float* wprojT = wqkvT + (size_t)C_ * C3_;            // [C][C]

  // 0) transpose weights (one-time per call, 4 MB total)
  kT_transpose<<<dim3(C_ / 32, C3_ / 32), dim3(32, 8), 0, stream>>>(qkv_w,  wqkvT,  C3_, C_);
  kT_transpose<<<dim3(C_ / 32, C_  / 32), dim3(32, 8), 0, stream>>>(proj_w, wprojT, C_,  C_);

  // 1) GroupNorm
  kT_gn_stats<<<B_ * GRP_, 256, 0, stream>>>(x, mean, rstd);
  kT_gn_apply<<<B_ * C_,   256, 0, stream>>>(x, gn_w, gn_b, mean, rstd, xn);

  // 2) qkv = xn^T @ qkv_w^T + qkv_b   -> [B][N][3C]
  kT_gemm<N_, C3_, C3_, C_, false, true, false>
      <<<dim3(N_ / 64, C3_ / 256, B_), 256, 0, stream>>>(
      xn, (long long)C_ * N_,
      wqkvT, 0,
      qkv, (long long)N_ * C3_,
      qkv_b, nullptr, 0);

  // 3) k softmax in place (contiguous rows of 512)
  kT_softmax_k<<<B_ * N_, 256, 0, stream>>>(qkv);

  // 4) q: tiled transpose to qT[b][d][n], then contiguous row softmax
  kT_qtranspose<<<dim3(C_ / 32, N_ / 32, B_), dim3(32, 8), 0, stream>>>(qkv, qT);
  kT_softmax_row1024<<<B_ * C_, 256, 0, stream>>>(qT);

  // 5) context[c][d] = sum_n k[n][c] * v[n][d]   -> [B][C][C]
  kT_gemm<C3_, C3_, C_, N_, false, false, false>
      <<<dim3(C_ / 64, C_ / 256, B_), 256, 0, stream>>>(
      qkv + C_,     (long long)N_ * C3_,   // A = k (m=c contiguous)
      qkv + 2 * C_, (long long)N_ * C3_,   // B = v
      ctx, (long long)C_ * C_,
      nullptr, nullptr, 0);

  // 6) out2[n][d] = sum_c q[n][c] * ctx[c][d]  (store transposed -> [B][C][N])
  kT_gemm<N_, C_, N_, C_, true, false, false>
      <<<dim3(N_ / 64, C_ / 256, B_), 256, 0, stream>>>(
      qT, (long long)C_ * N_,
      ctx, (long long)C_ * C_,
      o2T, (long long)C_ * N_,
      nullptr, nullptr, 0);

  // 7) final = out2 @ proj_w^T + proj_b + xn  (transposed store -> [B][C][H][W])
  kT_gemm<N_, C_, N_, C_, true, true, true>
      <<<dim3(N_ / 64, C_ / 256, B_), 256, 0, stream>>>(
      o2T, (long long)C_ * N_,
      wprojT, 0,
      out, (long long)C_ * N_,
      proj_b, xn, (long long)C_ * N_);
}